// KMaxConcPooling_26482768347140
// MI455X (gfx1250) — compile-verified
//
#include <hip/hip_runtime.h>

#define THREADS    512
#define N_ITEMS    10000
#define D_FEAT     512
#define MAX_K      32
#define PER_THREAD 20    // ceil(10000 / 512)
#define NEG_INF    (-__builtin_inff())

#if defined(__has_builtin)
#  if __has_builtin(__builtin_amdgcn_global_load_async_to_lds_b128) && \
      __has_builtin(__builtin_amdgcn_s_wait_asynccnt)
#    define USE_ASYNC_LDS 1
#  else
#    define USE_ASYNC_LDS 0
#  endif
#else
#  define USE_ASYNC_LDS 0
#endif

typedef int v4i __attribute__((ext_vector_type(4)));

__global__ __launch_bounds__(THREADS)
void topk_gather_kernel(const float* __restrict__ s,
                        const float* __restrict__ x,
                        const int*   __restrict__ kptr,
                        float*       __restrict__ out)
{
    const int tid = threadIdx.x;
    const int b   = blockIdx.x;
    const int k   = min(kptr[0], MAX_K);

    __shared__ float rv[THREADS];
    __shared__ int   ri[THREADS];
    __shared__ float win_val[MAX_K];
    __shared__ int   win_idx[MAX_K];
#if USE_ASYNC_LDS
    __shared__ __align__(16) float stage[4 * D_FEAT];   // 4 rows x 2KB = 8KB
#endif

    // ---- load this thread's strided score slice into VGPRs (coalesced) ----
    const float* sb = s + (size_t)b * N_ITEMS;
    float lv[PER_THREAD];
#pragma unroll
    for (int i = 0; i < PER_THREAD; ++i) {
        int g = i * THREADS + tid;
        lv[i] = (g < N_ITEMS) ? sb[g] : NEG_INF;
    }

    const float* xb = x + (size_t)b * N_ITEMS * D_FEAT;

    // ---- iterative top-k: k rounds of block-wide argmax ----
    for (int it = 0; it < k; ++it) {
        // local argmax; ascending scan + strict '>' keeps the smallest index
        float bv = lv[0];
        int   bi = tid;
#pragma unroll
        for (int i = 1; i < PER_THREAD; ++i) {
            if (lv[i] > bv) { bv = lv[i]; bi = i * THREADS + tid; }
        }
        rv[tid] = bv; ri[tid] = bi;
        __syncthreads();
        for (int st = THREADS / 2; st > 0; st >>= 1) {
            if (tid < st) {
                float v2 = rv[tid + st];
                int   i2 = ri[tid + st];
                if (v2 > rv[tid] || (v2 == rv[tid] && i2 < ri[tid])) {
                    rv[tid] = v2; ri[tid] = i2;
                }
            }
            __syncthreads();
        }
        const int   widx = ri[0];
        const float wv   = rv[0];
        __syncthreads();  // all threads consumed winner before rv/ri reuse

        if (tid == 0) { win_val[it] = wv; win_idx[it] = widx; }
        // invalidate winner in its owner's registers
        if ((widx & (THREADS - 1)) == tid) lv[widx >> 9] = NEG_INF;
        // prefetch the winning 2KB feature row (16 lanes x 128B) so it is
        // resident by gather time; emits global_prefetch_b8 on gfx1250
        if (tid < 16) {
            const char* rowp = (const char*)(xb + (size_t)widx * D_FEAT);
            __builtin_prefetch(rowp + tid * 128, 0, 0);
        }
    }
    __syncthreads();

    // ---- gather k rows of 512 floats; 4 rows per round, 128 lanes/row ----
    const int group = tid >> 7;    // 0..3 : which row of this round
    const int lane  = tid & 127;   // 16B chunk within the row
    for (int r0 = 0; r0 < k; r0 += 4) {
        const int  j      = r0 + group;
        const bool active = (j < k);
        const int  row    = active ? win_idx[j] : 0;
        const float* src  = xb + (size_t)row * D_FEAT + lane * 4;

        float4 v;
#if USE_ASYNC_LDS
        if (active) {
            // per-lane async DMA: global (16B aligned) -> LDS, no VGPR data
            __builtin_amdgcn_global_load_async_to_lds_b128(
                (v4i*)(void*)src,
                (v4i*)(void*)&stage[tid * 4],
                0, 0);
        }
        __builtin_amdgcn_s_wait_asynccnt(0);
        __syncthreads();
        v = *(const float4*)&stage[tid * 4];
#else
        v = active ? *(const float4*)src : make_float4(0.f, 0.f, 0.f, 0.f);
#endif
        if (active) {
            // output row starts at +1 float => 4B-aligned only: scalar stores
            float* dst = out + ((size_t)b * k + j) * (D_FEAT + 1) + 1 + lane * 4;
            dst[0] = v.x; dst[1] = v.y; dst[2] = v.z; dst[3] = v.w;
            if (lane == 0) dst[-1] = win_val[j];
        }
#if USE_ASYNC_LDS
        __syncthreads();  // staging buffer reuse next round
#endif
    }
}

extern "C" void kernel_launch(void* const* d_in, const int* in_sizes, int n_in,
                              void* d_out, int out_size, void* d_ws, size_t ws_size,
                              hipStream_t stream) {
    const float* s    = (const float*)d_in[0];
    const float* x    = (const float*)d_in[1];
    const int*   kptr = (const int*)d_in[2];
    float*       out  = (float*)d_out;

    const int bs = in_sizes[0] / N_ITEMS;   // s is [bs, n, 1]
    topk_gather_kernel<<<bs, THREADS, 0, stream>>>(s, x, kptr, out);
}